// GNNLayer_41171556499662
// MI455X (gfx1250) — compile-verified
//
#include <hip/hip_runtime.h>
#include <math.h>

#define NN 50000
#define NE 800000
#define D 64
#define SLOPE 0.01f

typedef float v2f __attribute__((ext_vector_type(2)));
typedef float v8f __attribute__((ext_vector_type(8)));

__global__ void fill_kernel(float* __restrict__ p, float v, int n) {
  int i = blockIdx.x * blockDim.x + threadIdx.x;
  if (i < n) p[i] = v;
}

__global__ void degree_kernel(const int* __restrict__ src, const int* __restrict__ dst,
                              int* __restrict__ deg_s, int* __restrict__ deg_d) {
  int e = blockIdx.x * blockDim.x + threadIdx.x;
  if (e < NE) {
    atomicAdd(&deg_s[src[e]], 1);
    atomicAdd(&deg_d[dst[e]], 1);
  }
}

// z = h @ W_fc^T using V_WMMA_F32_16X16X4_F32.
// One wave computes a 16(row)x64(col) strip; 16 k-steps x 4 col tiles = 64 WMMAs.
__global__ void gemm_z_kernel(const float* __restrict__ h, const float* __restrict__ Wfc,
                              float* __restrict__ z) {
  const int lane = threadIdx.x & 31;
  const int wave = threadIdx.x >> 5;
  const int tile = blockIdx.x * 4 + wave;           // 16-row tile index
  if (tile * 16 >= NN) return;
  const int r  = lane & 15;                          // row-in-tile (A) / col-in-tile (B/D)
  const int kh = (lane >> 4) * 2;                    // K sub-offset: lanes 16-31 hold K=2,3
  const float* hrow = h + (size_t)(tile * 16 + r) * D;

  v8f acc[4];
  #pragma unroll
  for (int c = 0; c < 4; ++c)
    #pragma unroll
    for (int j = 0; j < 8; ++j) acc[c][j] = 0.0f;

  #pragma unroll
  for (int k = 0; k < D; k += 4) {
    v2f a = *(const v2f*)(hrow + k + kh);            // A: 16x4 fp32 fragment
    #pragma unroll
    for (int c = 0; c < 4; ++c) {
      // B[k', n] = W_fc[n, k']  (z = h @ W_fc^T)
      v2f b = *(const v2f*)(Wfc + (size_t)(c * 16 + r) * D + k + kh);
      acc[c] = __builtin_amdgcn_wmma_f32_16x16x4_f32(
          false, a, false, b, (short)0, acc[c], false, false);
    }
  }

  const int mhi = (lane >> 4) * 8;                   // D layout: lanes 16-31 hold M=8..15
  #pragma unroll
  for (int c = 0; c < 4; ++c)
    #pragma unroll
    for (int j = 0; j < 8; ++j)
      z[(size_t)(tile * 16 + mhi + j) * D + c * 16 + r] = acc[c][j];
}

// Per-node: degree-weighted raw moment sums (S1,S2,S3 for src & dst sides) and
// attention projections p[n]=z[n].blk0, q[n]=z[n].blk3. One wave per node.
__global__ void node_stats_kernel(const float* __restrict__ z, const int* __restrict__ degs,
                                  const int* __restrict__ degd, const float* __restrict__ wa,
                                  float* __restrict__ p, float* __restrict__ q,
                                  float* __restrict__ sums) {
  __shared__ float S[6 * D];
  const int tid = threadIdx.x;
  for (int i = tid; i < 6 * D; i += blockDim.x) S[i] = 0.0f;
  __syncthreads();

  const int lane = tid & 31;
  const int node = blockIdx.x * 8 + (tid >> 5);
  if (node < NN) {
    const float wS = (float)degs[node];
    const float wD = (float)degd[node];
    float ps = 0.0f, qs = 0.0f;
    #pragma unroll
    for (int t = 0; t < 2; ++t) {
      const int f = lane + 32 * t;
      const float zv = z[(size_t)node * D + f];
      const float z2 = zv * zv, z3 = z2 * zv;
      atomicAdd(&S[f],         wS * zv);
      atomicAdd(&S[D + f],     wS * z2);
      atomicAdd(&S[2 * D + f], wS * z3);
      atomicAdd(&S[3 * D + f], wD * zv);
      atomicAdd(&S[4 * D + f], wD * z2);
      atomicAdd(&S[5 * D + f], wD * z3);
      ps += zv * wa[f];            // blk0
      qs += zv * wa[3 * D + f];    // blk3 (= blk[K])
    }
    #pragma unroll
    for (int off = 16; off > 0; off >>= 1) {
      ps += __shfl_xor(ps, off, 32);
      qs += __shfl_xor(qs, off, 32);
    }
    if (lane == 0) { p[node] = ps; q[node] = qs; }
  }
  __syncthreads();
  for (int i = tid; i < 6 * D; i += blockDim.x) atomicAdd(&sums[i], S[i]);
}

// Central moments from raw sums, signed roots, project through W_m1/W_m2,
// dot with attention blocks 1,2,4,5 -> single scalar a_const.
__global__ void finalize_kernel(const float* __restrict__ sums, const float* __restrict__ wa,
                                const float* __restrict__ Wm1, const float* __restrict__ Wm2,
                                float* __restrict__ aconst) {
  __shared__ float m2s[D], m3s[D], m2d[D], m3d[D], red[D];
  const int j = threadIdx.x;                 // 0..63
  const float inv = 1.0f / (float)NE;
  {
    float mu = sums[j] * inv;
    float s2 = sums[D + j] * inv;
    float s3 = sums[2 * D + j] * inv;
    float c2 = s2 - mu * mu;
    float c3 = s3 - 3.0f * mu * s2 + 2.0f * mu * mu * mu;
    m2s[j] = copysignf(sqrtf(fabsf(c2)), c2);
    m3s[j] = copysignf(cbrtf(fabsf(c3)), c3);
  }
  {
    float mu = sums[3 * D + j] * inv;
    float s2 = sums[4 * D + j] * inv;
    float s3 = sums[5 * D + j] * inv;
    float c2 = s2 - mu * mu;
    float c3 = s3 - 3.0f * mu * s2 + 2.0f * mu * mu * mu;
    m2d[j] = copysignf(sqrtf(fabsf(c2)), c2);
    m3d[j] = copysignf(cbrtf(fabsf(c3)), c3);
  }
  __syncthreads();
  float sm2 = 0.0f, sm3 = 0.0f, dm2 = 0.0f, dm3 = 0.0f;
  for (int i = 0; i < D; ++i) {
    sm2 += m2s[i] * Wm1[j * D + i];
    sm3 += m3s[i] * Wm2[j * D + i];
    dm2 += m2d[i] * Wm1[j * D + i];
    dm3 += m3d[i] * Wm2[j * D + i];
  }
  red[j] = sm2 * wa[D + j] + sm3 * wa[2 * D + j] + dm2 * wa[4 * D + j] + dm3 * wa[5 * D + j];
  __syncthreads();
  if (j == 0) {
    float s = 0.0f;
    for (int i = 0; i < D; ++i) s += red[i];
    *aconst = s;
  }
}

__device__ __forceinline__ void atomicMaxF32(float* addr, float val) {
  int* ia = (int*)addr;
  int old = *ia;
  while (__int_as_float(old) < val) {
    int assumed = old;
    old = atomicCAS(ia, assumed, __float_as_int(val));
    if (old == assumed) break;
  }
}

__global__ void edge_att_kernel(const float* __restrict__ e, const int* __restrict__ src,
                                const int* __restrict__ dst, const float* __restrict__ Wedge,
                                const float* __restrict__ wa, const float* __restrict__ p,
                                const float* __restrict__ q, const float* __restrict__ aconst,
                                float* __restrict__ att, float* __restrict__ m) {
  int i = blockIdx.x * blockDim.x + threadIdx.x;
  if (i >= NE) return;
  float e0 = e[2 * i], e1 = e[2 * i + 1];
  float ex0 = e0 * Wedge[0] + e1 * Wedge[1];   // ex = e @ W_edge^T
  float ex1 = e0 * Wedge[2] + e1 * Wedge[3];
  float a = p[src[i]] + q[dst[i]] + ex0 * wa[6 * D] + ex1 * wa[6 * D + 1] + aconst[0];
  float av = a > 0.0f ? a : SLOPE * a;
  att[i] = av;
  atomicMaxF32(&m[dst[i]], av);
}

__global__ void edge_denom_kernel(const int* __restrict__ dst, const float* __restrict__ att,
                                  const float* __restrict__ m, float* __restrict__ denom) {
  int i = blockIdx.x * blockDim.x + threadIdx.x;
  if (i >= NE) return;
  int d = dst[i];
  atomicAdd(&denom[d], expf(att[i] - m[d]));
}

// out[dst] += w * (z[src] + ez); 64 lanes per edge (feature-parallel, coalesced).
__global__ void edge_msg_kernel(const float* __restrict__ z, const float* __restrict__ e,
                                const int* __restrict__ src, const int* __restrict__ dst,
                                const float* __restrict__ Wedge, const float* __restrict__ We2n,
                                const float* __restrict__ att, const float* __restrict__ m,
                                const float* __restrict__ denom, float* __restrict__ out) {
  const int tid = threadIdx.x;
  const int f = tid & 63;
  const int i = blockIdx.x * 4 + (tid >> 6);
  if (i >= NE) return;
  const int s = src[i], d = dst[i];
  float e0 = e[2 * i], e1 = e[2 * i + 1];
  float ex0 = e0 * Wedge[0] + e1 * Wedge[1];
  float ex1 = e0 * Wedge[2] + e1 * Wedge[3];
  float ez = We2n[f * 2] * ex0 + We2n[f * 2 + 1] * ex1;  // W_e2n is (64,2)
  float w = expf(att[i] - m[d]) / denom[d];
  atomicAdd(&out[(size_t)d * D + f], w * (z[(size_t)s * D + f] + ez));
}

extern "C" void kernel_launch(void* const* d_in, const int* in_sizes, int n_in,
                              void* d_out, int out_size, void* d_ws, size_t ws_size,
                              hipStream_t stream) {
  const float* h     = (const float*)d_in[0];
  const float* e     = (const float*)d_in[1];
  const int*   src   = (const int*)d_in[2];
  const int*   dst   = (const int*)d_in[3];
  const float* Wfc   = (const float*)d_in[4];
  const float* Wattn = (const float*)d_in[5];
  const float* Wedge = (const float*)d_in[6];
  const float* We2n  = (const float*)d_in[7];
  const float* Wm1   = (const float*)d_in[8];
  const float* Wm2   = (const float*)d_in[9];
  float* out = (float*)d_out;

  float* ws = (float*)d_ws;
  size_t off = 0;
  float* z      = ws + off; off += (size_t)NN * D;
  float* p      = ws + off; off += NN;
  float* q      = ws + off; off += NN;
  float* m      = ws + off; off += NN;
  float* denom  = ws + off; off += NN;
  float* att    = ws + off; off += NE;
  float* sums   = ws + off; off += 6 * D;
  float* aconst = ws + off; off += 1;
  int* deg_s = (int*)(ws + off); off += NN;
  int* deg_d = (int*)(ws + off); off += NN;

  // init (harness poisons d_out/d_ws; graph replays re-run these fills)
  fill_kernel<<<(NN * D + 255) / 256, 256, 0, stream>>>(out, 0.0f, NN * D);
  fill_kernel<<<(2 * NN + 255) / 256, 256, 0, stream>>>((float*)deg_s, 0.0f, 2 * NN);
  fill_kernel<<<(NN + 255) / 256, 256, 0, stream>>>(m, -INFINITY, NN);
  fill_kernel<<<(NN + 255) / 256, 256, 0, stream>>>(denom, 0.0f, NN);
  fill_kernel<<<(6 * D + 1 + 255) / 256, 256, 0, stream>>>(sums, 0.0f, 6 * D + 1);

  degree_kernel<<<(NE + 255) / 256, 256, 0, stream>>>(src, dst, deg_s, deg_d);

  const int ntiles = (NN + 15) / 16;   // 3125
  gemm_z_kernel<<<(ntiles + 3) / 4, 128, 0, stream>>>(h, Wfc, z);

  node_stats_kernel<<<(NN + 7) / 8, 256, 0, stream>>>(z, deg_s, deg_d, Wattn, p, q, sums);

  finalize_kernel<<<1, 64, 0, stream>>>(sums, Wattn, Wm1, Wm2, aconst);

  edge_att_kernel<<<(NE + 255) / 256, 256, 0, stream>>>(e, src, dst, Wedge, Wattn, p, q,
                                                        aconst, att, m);

  edge_denom_kernel<<<(NE + 255) / 256, 256, 0, stream>>>(dst, att, m, denom);

  edge_msg_kernel<<<(NE + 3) / 4, 256, 0, stream>>>(z, e, src, dst, Wedge, We2n, att, m,
                                                    denom, out);
}